// Net_36146444763195
// MI455X (gfx1250) — compile-verified
//
#include <hip/hip_runtime.h>

typedef float v2f __attribute__((ext_vector_type(2)));
typedef float v8f __attribute__((ext_vector_type(8)));

// ---------------------------------------------------------------------------
// degree / dinv
// ---------------------------------------------------------------------------
__global__ void fill_ones_k(float* __restrict__ p, int n) {
  int i = blockIdx.x * blockDim.x + threadIdx.x;
  if (i < n) p[i] = 1.0f;
}

__global__ void degree_k(const long long* __restrict__ ei, int E,
                         float* __restrict__ deg) {
  int e = blockIdx.x * blockDim.x + threadIdx.x;
  if (e < E) {
    int d = (int)ei[(size_t)E + e];
    atomicAdd(&deg[d], 1.0f);
  }
}

__global__ void rsqrt_k(float* __restrict__ p, int n) {
  int i = blockIdx.x * blockDim.x + threadIdx.x;
  if (i < n) p[i] = rsqrtf(p[i]);
}

// ---------------------------------------------------------------------------
// FP32 WMMA GEMM: C[M x N] = A[M x K] * B[K x N]
// one 16x16 C tile per wave; blockDim.x = (N/16)*32; grid.x = ceil(M/16)
// V_WMMA_F32_16X16X4_F32 lane layout:
//   A (16x4): lane L holds A[m0 + L%16][k + 2*(L/16) + v], v = 0,1  -> v2f
//   B (4x16): lane L holds B[k + 2*(L/16) + v][n0 + L%16]           -> v2f
//   C (16x16): VGPR v of lane L = C[m0 + v + 8*(L/16)][n0 + L%16]   -> v8f
// ---------------------------------------------------------------------------
template <int K, int N>
__global__ __launch_bounds__(256) void gemm_wmma_f32(
    const float* __restrict__ A, const float* __restrict__ B,
    float* __restrict__ C, int M) {
  const int lane = threadIdx.x & 31;
  const int wave = threadIdx.x >> 5;
  const int g = lane >> 4;   // K-pair group (0 or 1)
  const int r = lane & 15;   // row (A) / column (B,C) within tile
  const int m0 = blockIdx.x * 16;
  const int n0 = wave * 16;

  int arow = m0 + r;
  if (arow >= M) arow = M - 1;  // clamp (uniform tail; stores are guarded)
  const float* Ap = A + (size_t)arow * K + 2 * g;        // A[m][k+2g .. k+2g+1]
  const float* Bp = B + (size_t)(2 * g) * N + n0 + r;    // B[k+2g][n], B[k+2g+1][n]

  v8f acc = {};
#pragma unroll 8
  for (int k = 0; k < K; k += 4) {
    v2f a = *(const v2f*)(Ap + k);  // 8B-aligned: K%4==0, rows 4*K-aligned
    v2f b;
    b.x = Bp[(size_t)k * N];
    b.y = Bp[(size_t)k * N + N];
    acc = __builtin_amdgcn_wmma_f32_16x16x4_f32(
        /*neg_a=*/false, a, /*neg_b=*/false, b,
        /*c_mod=*/(short)0, acc, /*reuse_a=*/false, /*reuse_b=*/false);
  }

  const int crow0 = m0 + 8 * g;
  float* Cp = C + (size_t)crow0 * N + n0 + r;
  if (m0 + 16 <= M) {
    // full tile (always the case when M % 16 == 0): unconditional stores,
    // no EXEC-mask manipulation
#pragma unroll
    for (int v = 0; v < 8; ++v) Cp[(size_t)v * N] = acc[v];
  } else {
#pragma unroll
    for (int v = 0; v < 8; ++v) {
      if (crow0 + v < M) Cp[(size_t)v * N] = acc[v];
    }
  }
}

// ---------------------------------------------------------------------------
// hs[i,f] = t[i,f] = dinv[i] * h[i,f]   (t init includes the self-loop term)
// ---------------------------------------------------------------------------
template <int F>
__global__ void scale_init_k(const float* __restrict__ h,
                             const float* __restrict__ dinv,
                             float* __restrict__ hs, float* __restrict__ t,
                             int n) {
  long long idx = (long long)blockIdx.x * blockDim.x + threadIdx.x;
  if (idx >= (long long)n * F) return;
  int i = (int)(idx / F);  // F is a power of two -> shift
  float v = dinv[i] * h[idx];
  hs[idx] = v;
  t[idx] = v;
}

// ---------------------------------------------------------------------------
// t[dst,:] += hs[src,:]  -- one wave per edge, coalesced 32-wide atomics
// ---------------------------------------------------------------------------
template <int F>
__global__ __launch_bounds__(256) void aggregate_k(
    const long long* __restrict__ ei, int E, const float* __restrict__ hs,
    float* __restrict__ t) {
  const int lane = threadIdx.x & 31;
  const int e = blockIdx.x * 8 + (threadIdx.x >> 5);
  if (e >= E) return;
  const int s = (int)ei[e];
  const int d = (int)ei[(size_t)E + e];
  const float* hrow = hs + (size_t)s * F;
  float* trow = t + (size_t)d * F;
#pragma unroll
  for (int f = lane; f < F; f += 32) atomicAdd(&trow[f], hrow[f]);
}

// ---------------------------------------------------------------------------
// h[i,f] = relu(dinv[i] * t[i,f] + b[f])
// ---------------------------------------------------------------------------
template <int F>
__global__ void finalize_relu_k(const float* __restrict__ t,
                                const float* __restrict__ dinv,
                                const float* __restrict__ b,
                                float* __restrict__ h, int n) {
  long long idx = (long long)blockIdx.x * blockDim.x + threadIdx.x;
  if (idx >= (long long)n * F) return;
  int i = (int)(idx / F);
  int f = (int)(idx - (long long)i * F);
  float v = dinv[i] * t[idx] + b[f];
  h[idx] = v > 0.0f ? v : 0.0f;
}

// ---------------------------------------------------------------------------
// out[i,:] = log_softmax(dinv[i]*t2[i,:] + b2)   -- one wave32 per row, F=64
// ---------------------------------------------------------------------------
__global__ __launch_bounds__(256) void logsoftmax_out_k(
    const float* __restrict__ t2, const float* __restrict__ dinv,
    const float* __restrict__ b, float* __restrict__ out, int n) {
  const int lane = threadIdx.x & 31;
  const int row = blockIdx.x * 8 + (threadIdx.x >> 5);
  if (row >= n) return;
  const float di = dinv[row];
  const float* tr = t2 + (size_t)row * 64;
  float v0 = di * tr[lane] + b[lane];
  float v1 = di * tr[lane + 32] + b[lane + 32];

  float m = fmaxf(v0, v1);
#pragma unroll
  for (int off = 16; off >= 1; off >>= 1) m = fmaxf(m, __shfl_xor(m, off, 32));
  float s = expf(v0 - m) + expf(v1 - m);
#pragma unroll
  for (int off = 16; off >= 1; off >>= 1) s += __shfl_xor(s, off, 32);
  float lse = m + logf(s);

  float* orow = out + (size_t)row * 64;
  orow[lane] = v0 - lse;
  orow[lane + 32] = v1 - lse;
}

// ---------------------------------------------------------------------------
// launch
// ---------------------------------------------------------------------------
extern "C" void kernel_launch(void* const* d_in, const int* in_sizes, int n_in,
                              void* d_out, int out_size, void* d_ws,
                              size_t ws_size, hipStream_t stream) {
  const float* x = (const float*)d_in[0];            // [n, 512]
  const long long* ei = (const long long*)d_in[1];   // [2, E] int64
  const float* W1 = (const float*)d_in[2];           // [512, 128]
  const float* b1 = (const float*)d_in[3];           // [128]
  const float* W2 = (const float*)d_in[4];           // [128, 64]
  const float* b2 = (const float*)d_in[5];           // [64]
  float* out = (float*)d_out;                        // [n, 64]

  const int K1 = 512, F1 = 128, F2 = 64;
  const int n = in_sizes[0] / K1;    // 100000
  const int E = in_sizes[1] / 2;     // 1600000

  // workspace carve-up (bytes)
  char* ws = (char*)d_ws;
  const size_t REG = 52428800ull;  // 50 MiB per region (>= 100000*128*4)
  float* dinv = (float*)ws;                          // n floats
  float* bufh = (float*)(ws + (1ull << 20));         // h1   [n,128]
  float* bufhs = (float*)(ws + (1ull << 20) + REG);  // hs1  [n,128]
  float* buft = (float*)(ws + (1ull << 20) + 2 * REG);  // t1 [n,128]
  // layer-2 reuse (sequential, no overlap in time):
  float* h2 = bufhs;                                  // [n,64]
  float* hs2 = buft;                                  // [n,64]
  float* t2 = (float*)((char*)buft + 26214400ull);    // [n,64]

  const int mt = (n + 15) / 16;  // 6250 M-tiles

  // deg -> dinv
  fill_ones_k<<<(n + 255) / 256, 256, 0, stream>>>(dinv, n);
  degree_k<<<(E + 255) / 256, 256, 0, stream>>>(ei, E, dinv);
  rsqrt_k<<<(n + 255) / 256, 256, 0, stream>>>(dinv, n);

  // layer 1: h = x @ W1 ; hs = t = dinv*h ; t += scatter ; h = relu(dinv*t + b1)
  gemm_wmma_f32<512, 128><<<mt, 256, 0, stream>>>(x, W1, bufh, n);
  {
    long long tot = (long long)n * F1;
    int blocks = (int)((tot + 255) / 256);
    scale_init_k<128><<<blocks, 256, 0, stream>>>(bufh, dinv, bufhs, buft, n);
    aggregate_k<128><<<(E + 7) / 8, 256, 0, stream>>>(ei, E, bufhs, buft);
    finalize_relu_k<128><<<blocks, 256, 0, stream>>>(buft, dinv, b1, bufh, n);
  }

  // layer 2: h2 = h1 @ W2 ; hs2 = t2 = dinv*h2 ; t2 += scatter ; log_softmax
  gemm_wmma_f32<128, 64><<<mt, 128, 0, stream>>>(bufh, W2, h2, n);
  {
    long long tot = (long long)n * F2;
    int blocks = (int)((tot + 255) / 256);
    scale_init_k<64><<<blocks, 256, 0, stream>>>(h2, dinv, hs2, t2, n);
    aggregate_k<64><<<(E + 7) / 8, 256, 0, stream>>>(ei, E, hs2, t2);
    logsoftmax_out_k<<<(n + 7) / 8, 256, 0, stream>>>(t2, dinv, b2, out, n);
  }
}